// EnhancedGAT_67645734912986
// MI455X (gfx1250) — compile-verified
//
#include <hip/hip_runtime.h>

#define NN 50000
#define EE 400000
#define GG 512
#define FF 128
#define KK 4
#define HH 64
#define KH 256
#define BN_EPS 1e-5f
#define KCHUNK 128

typedef float v2f __attribute__((ext_vector_type(2)));
typedef float v8f __attribute__((ext_vector_type(8)));
typedef int   v4i __attribute__((ext_vector_type(4)));
typedef __attribute__((address_space(1))) v4i* gv4i_ptr;   // global int4*
typedef __attribute__((address_space(3))) v4i* lv4i_ptr;   // LDS int4*

__device__ __forceinline__ float lrelu02(float x) { return x > 0.f ? x : 0.2f * x; }
__device__ __forceinline__ float sigmoidf_(float x) { return 1.f / (1.f + __expf(-x)); }

// Float atomic max via sign-split int/uint atomics (portable, still global atomics).
__device__ __forceinline__ void atomicMaxF(float* addr, float v) {
    if (v >= 0.f) atomicMax((int*)addr, __float_as_int(v));
    else          atomicMin((unsigned int*)addr, __float_as_uint(v));
}

#if __has_builtin(__builtin_amdgcn_global_load_async_to_lds_b128)
#define HAVE_ASYNC_LDS 1
#else
#define HAVE_ASYNC_LDS 0
#endif

__device__ __forceinline__ void wait_async_lds() {
#if HAVE_ASYNC_LDS
#if __has_builtin(__builtin_amdgcn_s_wait_asynccnt)
    __builtin_amdgcn_s_wait_asynccnt(0);
#else
    asm volatile("s_wait_asynccnt 0" ::: "memory");
#endif
#endif
}

// ---------------------------------------------------------------------------
// C[M,Ncols] = A[M,Kdim] @ B[Kdim,Ncols], all row-major, lda=Kdim, ldb=ldc=Ncols.
// Block: 128 threads (4 waves). Each wave -> 32 rows x 64 cols via 8 x
// V_WMMA_F32_16X16X4_F32 accumulators (B fragments reused across 2 row tiles).
// B panel staged in LDS in 128-K chunks via async global->LDS b128 copies.
// Requires: M % 16 == 0, Kdim % 4 == 0, Ncols % 64 == 0.
// ---------------------------------------------------------------------------
__global__ void __launch_bounds__(128)
gemm_wmma_f32(const float* __restrict__ A, const float* __restrict__ B,
              float* __restrict__ C, int M, int Kdim, int Ncols)
{
    __shared__ __attribute__((aligned(16))) float ldsB[KCHUNK * 64];   // 32 KB

    const int tid     = threadIdx.x;
    const int colBase = blockIdx.x * 64;
    const int wave    = tid >> 5;
    const int lane    = tid & 31;
    const int lid     = lane & 15;
    const int half    = lane >> 4;
    const int rowBase = blockIdx.y * 128 + wave * 32;
    const bool act0   = (rowBase < M);          // wave-uniform guards: EXEC all-ones
    const bool act1   = (rowBase + 16 < M);     // for every WMMA

    // A fragment pointers: A[row][k + 2*half + {0,1}]
    const float* Arow0 = A + (size_t)((act0 ? rowBase : 0) + lid) * Kdim + 2 * half;
    const float* Arow1 = A + (size_t)((act1 ? rowBase + 16 : 0) + lid) * Kdim + 2 * half;

    v8f c00 = {}, c01 = {}, c02 = {}, c03 = {};
    v8f c10 = {}, c11 = {}, c12 = {}, c13 = {};

    for (int kb = 0; kb < Kdim; kb += KCHUNK) {
        __syncthreads();
        const int kc = (Kdim - kb) < KCHUNK ? (Kdim - kb) : KCHUNK;
        // B panel fill: kc rows x 64 cols, 16B granules (16 per row)
        for (int i = tid; i < kc * 16; i += 128) {
            const int kr = i >> 4, q = (i & 15) << 2;
            const float* gp = B + (size_t)(kb + kr) * Ncols + colBase + q;
#if HAVE_ASYNC_LDS
            __builtin_amdgcn_global_load_async_to_lds_b128(
                (gv4i_ptr)gp, (lv4i_ptr)(ldsB + kr * 64 + q), 0, 0);
#else
            *(uint4*)(ldsB + kr * 64 + q) = *(const uint4*)gp;
#endif
        }
        wait_async_lds();
        __syncthreads();

        for (int k = 0; k < kc; k += 4) {
            v2f a0, a1;
            a0.x = Arow0[kb + k]; a0.y = Arow0[kb + k + 1];
            a1.x = Arow1[kb + k]; a1.y = Arow1[kb + k + 1];
            const int r0 = (k + 2 * half) * 64;   // B rows k+2*half (x) and +1 (y)
            v2f b;
            b.x = ldsB[r0 + lid];      b.y = ldsB[r0 + 64 + lid];
            c00 = __builtin_amdgcn_wmma_f32_16x16x4_f32(false, a0, false, b, (short)0, c00, false, false);
            c10 = __builtin_amdgcn_wmma_f32_16x16x4_f32(false, a1, false, b, (short)0, c10, false, false);
            b.x = ldsB[r0 + 16 + lid]; b.y = ldsB[r0 + 80 + lid];
            c01 = __builtin_amdgcn_wmma_f32_16x16x4_f32(false, a0, false, b, (short)0, c01, false, false);
            c11 = __builtin_amdgcn_wmma_f32_16x16x4_f32(false, a1, false, b, (short)0, c11, false, false);
            b.x = ldsB[r0 + 32 + lid]; b.y = ldsB[r0 + 96 + lid];
            c02 = __builtin_amdgcn_wmma_f32_16x16x4_f32(false, a0, false, b, (short)0, c02, false, false);
            c12 = __builtin_amdgcn_wmma_f32_16x16x4_f32(false, a1, false, b, (short)0, c12, false, false);
            b.x = ldsB[r0 + 48 + lid]; b.y = ldsB[r0 + 112 + lid];
            c03 = __builtin_amdgcn_wmma_f32_16x16x4_f32(false, a0, false, b, (short)0, c03, false, false);
            c13 = __builtin_amdgcn_wmma_f32_16x16x4_f32(false, a1, false, b, (short)0, c13, false, false);
        }
    }

    // C/D layout: VGPR i -> row (tileRow + 8*half + i), col = colBase + 16*c + lid
    if (act0) {
        const int row0 = rowBase + 8 * half;
#pragma unroll
        for (int i = 0; i < 8; ++i) {
            float* Cp = C + (size_t)(row0 + i) * Ncols + colBase + lid;
            Cp[0]  = c00[i];
            Cp[16] = c01[i];
            Cp[32] = c02[i];
            Cp[48] = c03[i];
        }
    }
    if (act1) {
        const int row0 = rowBase + 16 + 8 * half;
#pragma unroll
        for (int i = 0; i < 8; ++i) {
            float* Cp = C + (size_t)(row0 + i) * Ncols + colBase + lid;
            Cp[0]  = c10[i];
            Cp[16] = c11[i];
            Cp[32] = c12[i];
            Cp[48] = c13[i];
        }
    }
}

// ---------------------------------------------------------------------------
// Elementwise / graph kernels
// ---------------------------------------------------------------------------
__global__ void copy4_f32(const uint4* __restrict__ in, uint4* __restrict__ out, int n4) {
    int i = blockIdx.x * blockDim.x + threadIdx.x;
    if (i < n4) out[i] = in[i];
}

__global__ void init_softmax(float* __restrict__ m, float* __restrict__ s, int n) {
    int i = blockIdx.x * blockDim.x + threadIdx.x;
    if (i < n) { m[i] = -3.0e38f; s[i] = 0.f; }
}

// el[n,k] = sum_h Z[n,k*H+h]*al[k,h]; er likewise
__global__ void node_attn(const float* __restrict__ Z, const float* __restrict__ al,
                          const float* __restrict__ ar, float* __restrict__ el,
                          float* __restrict__ er) {
    int i = blockIdx.x * blockDim.x + threadIdx.x;
    if (i >= NN * KK) return;
    const int n = i >> 2, k = i & 3;
    const float* z  = Z + (size_t)n * KH + k * HH;
    const float* pl = al + k * HH;
    const float* pr = ar + k * HH;
    float sl = 0.f, sr = 0.f;
#pragma unroll 8
    for (int h = 0; h < HH; ++h) { float zv = z[h]; sl += zv * pl[h]; sr += zv * pr[h]; }
    el[i] = sl; er[i] = sr;
}

__global__ void edge_score_max(const int* __restrict__ src, const int* __restrict__ dst,
                               const float* __restrict__ el, const float* __restrict__ er,
                               float* __restrict__ esc, float* __restrict__ m) {
    int i = blockIdx.x * blockDim.x + threadIdx.x;
    if (i >= EE * KK) return;
    const int e = i >> 2, k = i & 3;
    const int sN = src[e], dN = dst[e];
    const float sc = lrelu02(el[sN * 4 + k] + er[dN * 4 + k]);
    esc[i] = sc;
    atomicMaxF(&m[dN * 4 + k], sc);
}

__global__ void edge_exp_sum(const int* __restrict__ dst, const float* __restrict__ m,
                             float* __restrict__ esc, float* __restrict__ s) {
    int i = blockIdx.x * blockDim.x + threadIdx.x;
    if (i >= EE * KK) return;
    const int e = i >> 2, k = i & 3;
    const int dN = dst[e];
    const float ee = __expf(esc[i] - m[dN * 4 + k]);
    esc[i] = ee;
    atomicAdd(&s[dN * 4 + k], ee);
}

// one block per edge; thread t -> feature t (k = t>>6). rst[dst] += a * Z[src]
__global__ void edge_scatter(const int* __restrict__ src, const int* __restrict__ dst,
                             const float* __restrict__ esc, const float* __restrict__ s,
                             const float* __restrict__ Z, float* __restrict__ rst) {
    const int e = blockIdx.x;
    const int t = threadIdx.x;
    const int k = t >> 6;
    const int sN = src[e], dN = dst[e];
    const float coeff = esc[e * 4 + k] / s[dN * 4 + k];
    atomicAdd(&rst[(size_t)dN * KH + t], coeff * Z[(size_t)sN * KH + t]);
}

__global__ void epilogue_flatten(float* __restrict__ rst, const float* __restrict__ b) {
    int i = blockIdx.x * blockDim.x + threadIdx.x;
    if (i >= NN * KH) return;
    const float v = rst[i] + b[i & 255];
    rst[i] = v > 0.f ? v : 0.f;
}

__global__ void epilogue_mean(const float* __restrict__ rst, const float* __restrict__ b,
                              float* __restrict__ x2) {
    int i = blockIdx.x * blockDim.x + threadIdx.x;
    if (i >= NN * HH) return;
    const int n = i >> 6, h = i & 63;
    float acc = 0.f;
#pragma unroll
    for (int k = 0; k < KK; ++k) {
        const float v = rst[(size_t)n * KH + k * HH + h] + b[k * HH + h];
        acc += v > 0.f ? v : 0.f;
    }
    x2[i] = 0.25f * acc;
}

__global__ void node_weight(const float* __restrict__ x2, const float* __restrict__ Wg,
                            const float* __restrict__ bg, float* __restrict__ w) {
    int n = blockIdx.x * blockDim.x + threadIdx.x;
    if (n >= NN) return;
    const float* x = x2 + (size_t)n * HH;
    float acc = bg[0];
#pragma unroll 8
    for (int h = 0; h < HH; ++h) acc += x[h] * Wg[h];
    w[n] = sigmoidf_(acc);
}

__global__ void init_readout(float* __restrict__ gs, float* __restrict__ gm) {
    int i = blockIdx.x * blockDim.x + threadIdx.x;
    if (i < GG * HH) { gs[i] = 0.f; gm[i] = -3.0e38f; }
}

__global__ void readout_scatter(const float* __restrict__ x2, const float* __restrict__ w,
                                const int* __restrict__ gid, float* __restrict__ gs,
                                float* __restrict__ gm) {
    int i = blockIdx.x * blockDim.x + threadIdx.x;
    if (i >= NN * HH) return;
    const int n = i >> 6, h = i & 63;
    const int g = gid[n];
    const float xv = x2[i];
    atomicAdd(&gs[g * HH + h], w[n] * xv);
    atomicMaxF(&gm[g * HH + h], xv);
}

__global__ void build_gf(const float* __restrict__ gs, const float* __restrict__ gm,
                         float* __restrict__ gf) {
    int i = blockIdx.x * blockDim.x + threadIdx.x;
    if (i >= GG * HH) return;
    const int g = i >> 6, h = i & 63;
    gf[g * 128 + h]      = gs[i];
    gf[g * 128 + 64 + h] = gm[i];
}

__global__ void mlp_bn(float* __restrict__ y, const float* __restrict__ bm1,
                       const float* __restrict__ bn_g, const float* __restrict__ bn_b,
                       const float* __restrict__ bn_m, const float* __restrict__ bn_v) {
    int i = blockIdx.x * blockDim.x + threadIdx.x;
    if (i >= GG * HH) return;
    const int h = i & 63;
    float v = y[i] + bm1[h];
    v = v > 0.f ? v : 0.f;
    y[i] = (v - bn_m[h]) * rsqrtf(bn_v[h] + BN_EPS) * bn_g[h] + bn_b[h];
}

__global__ void mlp_out(const float* __restrict__ y, const float* __restrict__ Wm2,
                        const float* __restrict__ bm2, float* __restrict__ out) {
    int g = blockIdx.x * blockDim.x + threadIdx.x;
    if (g >= GG) return;
    const float* yr = y + (size_t)g * HH;
    float acc = bm2[0];
#pragma unroll 8
    for (int h = 0; h < HH; ++h) acc += yr[h] * Wm2[h];
    out[g] = sigmoidf_(acc);
}

// ---------------------------------------------------------------------------
extern "C" void kernel_launch(void* const* d_in, const int* in_sizes, int n_in,
                              void* d_out, int out_size, void* d_ws, size_t ws_size,
                              hipStream_t stream) {
    (void)in_sizes; (void)n_in; (void)out_size; (void)ws_size;

    const float* h     = (const float*)d_in[0];
    const int*   srcp  = (const int*)  d_in[1];
    const int*   dstp  = (const int*)  d_in[2];
    const int*   gid   = (const int*)  d_in[3];
    const float* W0    = (const float*)d_in[4];
    const float* al0   = (const float*)d_in[5];
    const float* ar0   = (const float*)d_in[6];
    const float* b0    = (const float*)d_in[7];
    const float* resW0 = (const float*)d_in[8];
    const float* W1    = (const float*)d_in[9];
    const float* al1   = (const float*)d_in[10];
    const float* ar1   = (const float*)d_in[11];
    const float* b1    = (const float*)d_in[12];
    const float* W2    = (const float*)d_in[13];
    const float* al2   = (const float*)d_in[14];
    const float* ar2   = (const float*)d_in[15];
    const float* b2    = (const float*)d_in[16];
    const float* Wg    = (const float*)d_in[17];
    const float* bg    = (const float*)d_in[18];
    const float* Wm1   = (const float*)d_in[19];
    const float* bm1   = (const float*)d_in[20];
    const float* bn_g  = (const float*)d_in[21];
    const float* bn_b  = (const float*)d_in[22];
    const float* bn_m  = (const float*)d_in[23];
    const float* bn_v  = (const float*)d_in[24];
    const float* Wm2   = (const float*)d_in[25];
    const float* bm2   = (const float*)d_in[26];

    // workspace layout (floats)
    float* p = (float*)d_ws;
    float* bufA = p; p += (size_t)NN * KH;
    float* bufB = p; p += (size_t)NN * KH;
    float* Zb   = p; p += (size_t)NN * KH;
    float* el   = p; p += (size_t)NN * KK;
    float* er   = p; p += (size_t)NN * KK;
    float* mm   = p; p += (size_t)NN * KK;
    float* ssum = p; p += (size_t)NN * KK;
    float* esc  = p; p += (size_t)EE * KK;
    float* x2   = p; p += (size_t)NN * HH;
    float* wn   = p; p += (size_t)NN;
    float* gs   = p; p += (size_t)GG * HH;
    float* gm   = p; p += (size_t)GG * HH;
    float* gf   = p; p += (size_t)GG * 2 * HH;
    float* yb   = p; p += (size_t)GG * HH;

    auto gemm = [&](const float* A, const float* B, float* C, int M, int Kd, int Nc) {
        dim3 grid(Nc / 64, (M + 127) / 128);
        gemm_wmma_f32<<<grid, 128, 0, stream>>>(A, B, C, M, Kd, Nc);
    };

    auto run_layer = [&](const float* xin, int Din, const float* W, const float* resW,
                         const float* al, const float* ar, const float* bvec,
                         float* rst, bool meanAgg, float* meanOut) {
        gemm(xin, W, Zb, NN, Din, KH);                                    // Z = x @ W
        if (resW) gemm(xin, resW, rst, NN, Din, KH);                      // res = x @ resW
        else copy4_f32<<<(NN * KH / 4 + 255) / 256, 256, 0, stream>>>(
                 (const uint4*)xin, (uint4*)rst, NN * KH / 4);
        node_attn<<<(NN * KK + 255) / 256, 256, 0, stream>>>(Zb, al, ar, el, er);
        init_softmax<<<(NN * KK + 255) / 256, 256, 0, stream>>>(mm, ssum, NN * KK);
        edge_score_max<<<(EE * KK + 255) / 256, 256, 0, stream>>>(srcp, dstp, el, er, esc, mm);
        edge_exp_sum<<<(EE * KK + 255) / 256, 256, 0, stream>>>(dstp, mm, esc, ssum);
        edge_scatter<<<EE, 256, 0, stream>>>(srcp, dstp, esc, ssum, Zb, rst);
        if (meanAgg) epilogue_mean<<<(NN * HH + 255) / 256, 256, 0, stream>>>(rst, bvec, meanOut);
        else         epilogue_flatten<<<(NN * KH + 255) / 256, 256, 0, stream>>>(rst, bvec);
    };

    // GAT layer 0: h[N,128] -> bufA[N,256]  (learned residual)
    run_layer(h, FF, W0, resW0, al0, ar0, b0, bufA, false, nullptr);
    // GAT layer 1: bufA -> bufB (identity residual)
    run_layer(bufA, KH, W1, nullptr, al1, ar1, b1, bufB, false, nullptr);
    // GAT layer 2: bufB -> bufA (scratch) -> mean -> x2[N,64]
    run_layer(bufB, KH, W2, nullptr, al2, ar2, b2, bufA, true, x2);

    // WeightedSumAndMax readout
    node_weight<<<(NN + 255) / 256, 256, 0, stream>>>(x2, Wg, bg, wn);
    init_readout<<<(GG * HH + 255) / 256, 256, 0, stream>>>(gs, gm);
    readout_scatter<<<(NN * HH + 255) / 256, 256, 0, stream>>>(x2, wn, gid, gs, gm);
    build_gf<<<(GG * HH + 255) / 256, 256, 0, stream>>>(gs, gm, gf);

    // MLP predictor: y = BN(relu(gf @ Wm1 + bm1)); out = sigmoid(y @ Wm2 + bm2)
    gemm(gf, Wm1, yb, GG, 2 * HH, HH);
    mlp_bn<<<(GG * HH + 255) / 256, 256, 0, stream>>>(yb, bm1, bn_g, bn_b, bn_m, bn_v);
    mlp_out<<<(GG + 255) / 256, 256, 0, stream>>>(yb, Wm2, bm2, (float*)d_out);
}